// Attention_54408645705978
// MI455X (gfx1250) — compile-verified
//
#include <hip/hip_runtime.h>
#include <math.h>

typedef float v2f __attribute__((ext_vector_type(2)));
typedef float v8f __attribute__((ext_vector_type(8)));

#define B_      64
#define T_      2048
#define RNNDIM  1024
#define EMBDIM  512
#define ATTDIM  128
#define NFILT   32
#define KSZ     31
#define PADW    15
#define TCHUNK  16              // context-stage T split (concurrency for HBM)
#define TPER    (T_ / TCHUNK)   // 128 t-rows per partial block

// ---------------------------------------------------------------------------
// Kernel 1: processed_query[64,128] = query[64,1024] @ W_query[128,1024]^T
// One wave per 16x16 output tile; K-loop in steps of 4 using f32 WMMA.
// A 16x4 f32 layout (ISA 7.12.2): lane m=lane&15, k = k0 + (lane>>4)*2 (+1 in .y)
// ---------------------------------------------------------------------------
__global__ void pq_kernel(const float* __restrict__ query,
                          const float* __restrict__ Wq,
                          float* __restrict__ pq) {
    const int nt   = blockIdx.x;           // 0..7  (N tile)
    const int mt   = blockIdx.y;           // 0..3  (M tile)
    const int lane = threadIdx.x;
    const int half = lane >> 4;
    const int l16  = lane & 15;
    const int m = mt * 16 + l16;           // query row (batch index)
    const int n = nt * 16 + l16;           // output feature

    v8f acc = {};
    for (int k0 = 0; k0 < RNNDIM; k0 += 4) {
        const int k = k0 + half * 2;
        v2f a, b;
        a.x = query[m * RNNDIM + k];
        a.y = query[m * RNNDIM + k + 1];
        b.x = Wq[n * RNNDIM + k];          // B[k][n] = W_query[n][k]
        b.y = Wq[n * RNNDIM + k + 1];
        acc = __builtin_amdgcn_wmma_f32_16x16x4_f32(false, a, false, b,
                                                    (short)0, acc, false, false);
    }
    // C layout: VGPR r -> row mt*16 + r (lanes 0-15) / +8 (lanes 16-31), col = l16
    const int mrow = mt * 16 + half * 8;
#pragma unroll
    for (int r = 0; r < 8; ++r)
        pq[(mrow + r) * ATTDIM + nt * 16 + l16] = acc[r];
}

// ---------------------------------------------------------------------------
// Kernel 2: location conv (2 -> 32 channels, k=31, same padding), stored
// t-major as locT[B][T][32] so it is the A matrix of the next GEMM.
// One wave handles one t (32 lanes = 32 filters); conv-window reads broadcast.
// ---------------------------------------------------------------------------
__global__ void conv_kernel(const float* __restrict__ awc,   // [B,2,T]
                            const float* __restrict__ cw,    // [32,2,31]
                            float* __restrict__ locT) {      // [B,T,32]
    const int f = threadIdx.x;                       // 0..31
    const int t = blockIdx.x * 8 + threadIdx.y;      // 0..T-1
    const int b = blockIdx.y;
    float s = 0.f;
#pragma unroll
    for (int c = 0; c < 2; ++c) {
        const float* a = awc + (size_t)(b * 2 + c) * T_;
        const float* w = cw + (f * 2 + c) * KSZ;
#pragma unroll
        for (int k = 0; k < KSZ; ++k) {
            const int tp = t - PADW + k;
            const float av = (tp >= 0 && tp < T_) ? a[tp] : 0.f;
            s += av * w[k];
        }
    }
    locT[((size_t)b * T_ + t) * NFILT + f] = s;
}

// ---------------------------------------------------------------------------
// Kernel 3: per (b, 16-row t-tile):
//   PL(16x128) = locT_tile(16x32) @ W_loc(128x32)^T     via 8x8 f32 WMMAs
//   e[t] = sum_a v[a] * tanh(PL + pm + pq)              fused epilogue
// One wave per block (EXEC all ones as WMMA requires).
// ---------------------------------------------------------------------------
__global__ void energy_kernel(const float* __restrict__ locT,  // [B,T,32]
                              const float* __restrict__ Wloc,  // [128,32]
                              const float* __restrict__ pm,    // [B,T,128]
                              const float* __restrict__ pq,    // [B,128]
                              const float* __restrict__ vw,    // [128]
                              float* __restrict__ energies) {  // [B,T]
    const int tt   = blockIdx.x;          // t tile, 0..127
    const int b    = blockIdx.y;
    const int t0   = tt * 16;
    const int lane = threadIdx.x;
    const int half = lane >> 4;
    const int l16  = lane & 15;

    v8f acc[8];
#pragma unroll
    for (int i = 0; i < 8; ++i) acc[i] = (v8f){};

    const float* Arow = locT + ((size_t)b * T_ + t0 + l16) * NFILT;  // A row m=l16
#pragma unroll
    for (int k0 = 0; k0 < NFILT; k0 += 4) {
        const int k = k0 + half * 2;
        v2f a;
        a.x = Arow[k];
        a.y = Arow[k + 1];
#pragma unroll
        for (int nt = 0; nt < 8; ++nt) {
            const int n = nt * 16 + l16;
            v2f bb;
            bb.x = Wloc[n * NFILT + k];    // B[k][n] = W_loc[n][k]
            bb.y = Wloc[n * NFILT + k + 1];
            acc[nt] = __builtin_amdgcn_wmma_f32_16x16x4_f32(false, a, false, bb,
                                                            (short)0, acc[nt],
                                                            false, false);
        }
    }

    // Epilogue: rows of this half are mbase+r (r=0..7)
    const int mbase = t0 + half * 8;
    float e[8];
#pragma unroll
    for (int r = 0; r < 8; ++r) e[r] = 0.f;
#pragma unroll
    for (int r = 0; r < 8; ++r) {
        const float* pmrow = pm + ((size_t)b * T_ + mbase + r) * ATTDIM;
        const float* pqrow = pq + b * ATTDIM;
#pragma unroll
        for (int nt = 0; nt < 8; ++nt) {
            const int col = nt * 16 + l16;
            const float x = acc[nt][r] + pmrow[col] + pqrow[col];
            e[r] += vw[col] * tanhf(x);
        }
    }
    // reduce across the 16 lanes of each half (xor masks < 16 stay in-half)
#pragma unroll
    for (int r = 0; r < 8; ++r) {
        float s = e[r];
        s += __shfl_xor(s, 1, 32);
        s += __shfl_xor(s, 2, 32);
        s += __shfl_xor(s, 4, 32);
        s += __shfl_xor(s, 8, 32);
        e[r] = s;
    }
    if (l16 == 0) {  // lane 0 writes rows t0..t0+7, lane 16 rows t0+8..t0+15
#pragma unroll
        for (int r = 0; r < 8; ++r)
            energies[(size_t)b * T_ + mbase + r] = e[r];
    }
}

// ---------------------------------------------------------------------------
// Kernel 4: masked softmax over T per batch row. mask true -> -inf.
// ---------------------------------------------------------------------------
__global__ void softmax_kernel(const float* __restrict__ energies,
                               const unsigned char* __restrict__ mask,
                               float* __restrict__ wout) {
    const int b   = blockIdx.x;
    const int tid = threadIdx.x;    // 256 threads, 8 elems each
    __shared__ float red[256];

    float vals[8];
    float mx = -INFINITY;
#pragma unroll
    for (int i = 0; i < 8; ++i) {
        const int t = i * 256 + tid;
        const float v = mask[(size_t)b * T_ + t] ? -INFINITY
                                                 : energies[(size_t)b * T_ + t];
        vals[i] = v;
        mx = fmaxf(mx, v);
    }
    red[tid] = mx;
    __syncthreads();
    for (int s = 128; s > 0; s >>= 1) {
        if (tid < s) red[tid] = fmaxf(red[tid], red[tid + s]);
        __syncthreads();
    }
    mx = red[0];
    __syncthreads();

    float sum = 0.f;
#pragma unroll
    for (int i = 0; i < 8; ++i) {
        const float ev = (vals[i] == -INFINITY) ? 0.f : expf(vals[i] - mx);
        vals[i] = ev;
        sum += ev;
    }
    red[tid] = sum;
    __syncthreads();
    for (int s = 128; s > 0; s >>= 1) {
        if (tid < s) red[tid] += red[tid + s];
        __syncthreads();
    }
    const float inv = 1.f / red[0];
#pragma unroll
    for (int i = 0; i < 8; ++i) {
        const int t = i * 256 + tid;
        wout[(size_t)b * T_ + t] = vals[i] * inv;
    }
}

// ---------------------------------------------------------------------------
// Kernel 5a: partial context. The 256MB memory stream is the runtime floor
// (~11us at 23.3 TB/s), so split T into TCHUNK chunks per batch:
// 64*16 = 1024 blocks (4096 waves) of fully-coalesced float4 streaming —
// enough concurrency to cover HBM latency across all channels.
// partial[b][c][d] = sum_{t in chunk c} w[b,t] * memory[b,t,d]
// ---------------------------------------------------------------------------
__global__ void context_partial_kernel(const float* __restrict__ w,   // [B,T]
                                       const float* __restrict__ mem, // [B,T,512]
                                       float* __restrict__ partial) { // [B,16,512]
    const int c   = blockIdx.x;    // 0..TCHUNK-1
    const int b   = blockIdx.y;
    const int tid = threadIdx.x;   // 0..127 (float4 over 512)
    const int tbeg = c * TPER;
    const float4* m4 = (const float4*)(mem + (size_t)b * T_ * EMBDIM);
    float4 acc = {0.f, 0.f, 0.f, 0.f};
    for (int t = tbeg; t < tbeg + TPER; ++t) {
        const float wt = w[(size_t)b * T_ + t];
        const float4 v = m4[(size_t)t * (EMBDIM / 4) + tid];
        acc.x += wt * v.x;
        acc.y += wt * v.y;
        acc.z += wt * v.z;
        acc.w += wt * v.w;
    }
    ((float4*)(partial + ((size_t)b * TCHUNK + c) * EMBDIM))[tid] = acc;
}

// ---------------------------------------------------------------------------
// Kernel 5b: reduce the TCHUNK partials -> context[b,d]. Deterministic,
// no atomics (graph-capture safe, d_out poison irrelevant: full overwrite).
// ---------------------------------------------------------------------------
__global__ void context_reduce_kernel(const float* __restrict__ partial, // [B,16,512]
                                      float* __restrict__ ctx) {         // [B,512]
    const int b   = blockIdx.x;
    const int tid = threadIdx.x;   // 0..127
    float4 acc = {0.f, 0.f, 0.f, 0.f};
#pragma unroll
    for (int c = 0; c < TCHUNK; ++c) {
        const float4 v =
            ((const float4*)(partial + ((size_t)b * TCHUNK + c) * EMBDIM))[tid];
        acc.x += v.x;
        acc.y += v.y;
        acc.z += v.z;
        acc.w += v.w;
    }
    ((float4*)(ctx + (size_t)b * EMBDIM))[tid] = acc;
}

// ---------------------------------------------------------------------------
extern "C" void kernel_launch(void* const* d_in, const int* in_sizes, int n_in,
                              void* d_out, int out_size, void* d_ws, size_t ws_size,
                              hipStream_t stream) {
    (void)in_sizes; (void)n_in; (void)out_size; (void)ws_size;

    const float*         query  = (const float*)d_in[0];          // [64,1024]
    const float*         pm     = (const float*)d_in[1];          // [64,2048,128]
    const float*         awc    = (const float*)d_in[2];          // [64,2,2048]
    const unsigned char* mask   = (const unsigned char*)d_in[3];  // [64,2048] bool
    const float*         memory = (const float*)d_in[4];          // [64,2048,512]
    const float*         Wq     = (const float*)d_in[5];          // [128,1024]
    const float*         cw     = (const float*)d_in[6];          // [32,2,31]
    const float*         Wloc   = (const float*)d_in[7];          // [128,32]
    const float*         vw     = (const float*)d_in[8];          // [1,128]

    float* out  = (float*)d_out;
    float* ctx  = out;                       // [64,512]
    float* wout = out + B_ * EMBDIM;         // [64,2048]

    float* ws       = (float*)d_ws;
    float* pq       = ws;                                   // 64*128        (32 KB)
    float* locT     = pq + B_ * ATTDIM;                     // 64*2048*32    (16 MB)
    float* energies = locT + (size_t)B_ * T_ * NFILT;       // 64*2048       (512 KB)
    float* partial  = energies + (size_t)B_ * T_;           // 64*16*512     (2 MB)

    pq_kernel<<<dim3(ATTDIM / 16, B_ / 16), 32, 0, stream>>>(query, Wq, pq);
    conv_kernel<<<dim3(T_ / 8, B_), dim3(32, 8), 0, stream>>>(awc, cw, locT);
    energy_kernel<<<dim3(T_ / 16, B_), 32, 0, stream>>>(locT, Wloc, pm, pq, vw,
                                                        energies);
    softmax_kernel<<<B_, 256, 0, stream>>>(energies, mask, wout);
    context_partial_kernel<<<dim3(TCHUNK, B_), 128, 0, stream>>>(wout, memory,
                                                                 partial);
    context_reduce_kernel<<<B_, 128, 0, stream>>>(partial, ctx);
}